// tet10_densify_73572789780863
// MI455X (gfx1250) — compile-verified
//
#include <hip/hip_runtime.h>

typedef __attribute__((ext_vector_type(2))) float v2f;
typedef __attribute__((ext_vector_type(8))) float v8f;

#define NEG_SLOPE 0.01f

__device__ __forceinline__ float leaky(float x) { return fmaxf(x, NEG_SLOPE * x); }

// ---------------- LDS layout (units: floats) ----------------
// frag1: [2 branches][24 kchunks][4 ntiles][32 lanes][2 vals]  = 12288
// frag2: [2][16 kchunks][32 lanes][2 vals]                     = 2048
#define L_FRAG1 0
#define L_FRAG2 12288
#define L_B1    14336   // [2][64]
#define L_B2    14464   // [2][16]
#define L_W3    14496   // [2][4*16]
#define L_B3    14624   // [2][4]
#define L_W4    14632   // [2][2*4]
#define L_B4    14648   // [2][2]
#define L_W5    14652   // [2][2]
#define L_B5    14656   // [2]
#define L_OUT1  14720   // [8 waves][16][64] = 8192
#define L_OUT2  22912   // [8 waves][16][16] = 2048
#define L_TOTAL 24960
#define SMEM_BYTES (L_TOTAL * 4)

extern "C" __global__ __launch_bounds__(256)
void dual_mlp_wmma_kernel(const float* __restrict__ elems,
                          const float* __restrict__ enc,
                          const float* __restrict__ cw1, const float* __restrict__ cb1,
                          const float* __restrict__ cw2, const float* __restrict__ cb2,
                          const float* __restrict__ cw3, const float* __restrict__ cb3,
                          const float* __restrict__ cw4, const float* __restrict__ cb4,
                          const float* __restrict__ cw5, const float* __restrict__ cb5,
                          const float* __restrict__ tw1, const float* __restrict__ tb1,
                          const float* __restrict__ tw2, const float* __restrict__ tb2,
                          const float* __restrict__ tw3, const float* __restrict__ tb3,
                          const float* __restrict__ tw4, const float* __restrict__ tb4,
                          const float* __restrict__ tw5, const float* __restrict__ tb5,
                          float* __restrict__ out,
                          int rows, int tiles)
{
    extern __shared__ float lds[];
    const int tid  = threadIdx.x;
    const int lane = tid & 31;
    const int wave = tid >> 5;
    const int m    = lane & 15;   // row within 16-row tile (also N for B/C frags)
    const int kh   = lane >> 4;   // K-half selector within a 4-wide K chunk

    const float* w1b[2] = {cw1, tw1};
    const float* w2b[2] = {cw2, tw2};

    // ---------------- stage weight fragments into LDS ----------------
    // Logical x layout: cols [0,64) = enc, [64,94) = feat, [94,96) = zero pad.
    // frag value index: k = 4c + 2*(lane>=16) + v ; n = 16t + (lane&15)
    for (int i = tid; i < 12288; i += 256) {
        int v = i & 1;
        int l = (i >> 1) & 31;
        int t = (i >> 6) & 3;
        int r = i >> 8;               // br*24 + c
        int c = r % 24;
        int br = r / 24;
        int k = c * 4 + ((l >> 4) << 1) + v;
        int n = t * 16 + (l & 15);
        const float* w = w1b[br];
        float val = 0.f;
        if (k < 64)       val = w[n * 94 + 30 + k];   // enc region -> w cols 30..93
        else if (k < 94)  val = w[n * 94 + (k - 64)]; // feat region -> w cols 0..29
        lds[L_FRAG1 + i] = val;
    }
    for (int i = tid; i < 2048; i += 256) {
        int v = i & 1;
        int l = (i >> 1) & 31;
        int c = (i >> 6) & 15;
        int br = i >> 10;
        int k = c * 4 + ((l >> 4) << 1) + v;
        int n = l & 15;
        lds[L_FRAG2 + i] = w2b[br][n * 64 + k];
    }
    if (tid < 64) { lds[L_B1 + tid] = cb1[tid]; lds[L_B1 + 64 + tid] = tb1[tid];
                    lds[L_W3 + tid] = cw3[tid]; lds[L_W3 + 64 + tid] = tw3[tid]; }
    if (tid < 16) { lds[L_B2 + tid] = cb2[tid]; lds[L_B2 + 16 + tid] = tb2[tid]; }
    if (tid < 8)  { lds[L_W4 + tid] = cw4[tid]; lds[L_W4 + 8 + tid] = tw4[tid]; }
    if (tid < 4)  { lds[L_B3 + tid] = cb3[tid]; lds[L_B3 + 4 + tid] = tb3[tid]; }
    if (tid < 2)  { lds[L_B4 + tid] = cb4[tid]; lds[L_B4 + 2 + tid] = tb4[tid];
                    lds[L_W5 + tid] = cw5[tid]; lds[L_W5 + 2 + tid] = tw5[tid]; }
    if (tid == 0) { lds[L_B5] = cb5[0]; lds[L_B5 + 1] = tb5[0]; }
    __syncthreads();

    float* o1 = lds + L_OUT1 + wave * 1024;  // per-wave 16x64 staging
    float* o2 = lds + L_OUT2 + wave * 256;   // per-wave 16x16 staging

    const int stride = (int)gridDim.x * 8;
    const int bound  = ((tiles + stride - 1) / stride) * stride;  // uniform trip count

    for (int t0 = (int)blockIdx.x * 8 + wave; t0 < bound; t0 += stride) {
        const bool active = t0 < tiles;
        const int  row    = t0 * 16 + m;
        const bool rowok  = active && (row < rows);
        const int  srow   = rowok ? row : 0;

        const float* ep = enc   + (size_t)srow * 64;
        const float* fp = elems + (size_t)srow * 31;

        // Hoisted A fragments for layer 1 (shared by both branches).
        v2f afr[24];
        #pragma unroll
        for (int c = 0; c < 24; ++c) {
            const int kb = c * 4 + kh * 2;
            v2f a;
            if (c < 16) {                       // enc region (8B aligned)
                a = *(const v2f*)(ep + kb);
            } else if (c < 23) {                // feat region (4B aligned only)
                a.x = fp[kb - 64];
                a.y = fp[kb - 63];
            } else {                            // chunk 23: feat 28,29 then zero pad
                if (kh == 0) { a.x = fp[28]; a.y = fp[29]; }
                else         { a.x = 0.f;    a.y = 0.f;    }
            }
            afr[c] = a;
        }
        const float maskv = rowok ? fp[30] : 0.f;

        float res[2];
        #pragma unroll
        for (int br = 0; br < 2; ++br) {
            // ---- layer 1: [16x94(96)] @ [94x64] via 24 K-chunks x 4 N-tiles ----
            v8f acc[4] = {};
            #pragma unroll
            for (int c = 0; c < 24; ++c) {
                const float* fb = lds + L_FRAG1 + (br * 24 + c) * 256 + lane * 2;
                #pragma unroll
                for (int t = 0; t < 4; ++t) {
                    v2f b = *(const v2f*)(fb + t * 64);
                    acc[t] = __builtin_amdgcn_wmma_f32_16x16x4_f32(
                        false, afr[c], false, b, (short)0, acc[t], false, false);
                }
            }
            #pragma unroll
            for (int t = 0; t < 4; ++t) {
                const int n = t * 16 + m;
                const float bias = lds[L_B1 + br * 64 + n];
                #pragma unroll
                for (int v = 0; v < 8; ++v) {
                    float x = acc[t][v] + bias;
                    o1[(v + kh * 8) * 64 + n] = leaky(x);
                }
            }
            __syncthreads();

            // ---- layer 2: [16x64] @ [64x16] via 16 K-chunks ----
            v8f acc2 = {};
            #pragma unroll
            for (int c = 0; c < 16; ++c) {
                const int kb = c * 4 + kh * 2;
                v2f a2 = *(const v2f*)(o1 + m * 64 + kb);
                v2f b2 = *(const v2f*)(lds + L_FRAG2 + (br * 16 + c) * 64 + lane * 2);
                acc2 = __builtin_amdgcn_wmma_f32_16x16x4_f32(
                    false, a2, false, b2, (short)0, acc2, false, false);
            }
            {
                const float bias = lds[L_B2 + br * 16 + m];
                #pragma unroll
                for (int v = 0; v < 8; ++v) {
                    float x = acc2[v] + bias;
                    o2[(v + kh * 8) * 16 + m] = leaky(x);
                }
            }
            __syncthreads();

            // ---- layers 3..5 (16->4->2->1) per row in VALU ----
            float h2[16];
            #pragma unroll
            for (int n = 0; n < 16; ++n) h2[n] = o2[m * 16 + n];
            float h3[4];
            #pragma unroll
            for (int j = 0; j < 4; ++j) {
                float s = lds[L_B3 + br * 4 + j];
                #pragma unroll
                for (int n = 0; n < 16; ++n)
                    s = fmaf(h2[n], lds[L_W3 + br * 64 + j * 16 + n], s);
                h3[j] = leaky(s);
            }
            float h4[2];
            #pragma unroll
            for (int j = 0; j < 2; ++j) {
                float s = lds[L_B4 + br * 2 + j];
                #pragma unroll
                for (int i2 = 0; i2 < 4; ++i2)
                    s = fmaf(h3[i2], lds[L_W4 + br * 8 + j * 4 + i2], s);
                h4[j] = leaky(s);
            }
            float s5 = lds[L_B5 + br];
            s5 = fmaf(h4[0], lds[L_W5 + br * 2 + 0], s5);
            s5 = fmaf(h4[1], lds[L_W5 + br * 2 + 1], s5);
            res[br] = leaky(s5);
            __syncthreads();   // before o1/o2 reuse (next branch / next tile)
        }

        if (rowok && lane < 16) {
            const float sel = (maskv == 1.0f) ? res[0] : res[1];
            out[row] = fmaxf(sel, 0.f);
        }
    }
}

extern "C" void kernel_launch(void* const* d_in, const int* in_sizes, int n_in,
                              void* d_out, int out_size, void* d_ws, size_t ws_size,
                              hipStream_t stream) {
    (void)n_in; (void)out_size; (void)d_ws; (void)ws_size;
    const float* elems = (const float*)d_in[0];
    const float* enc   = (const float*)d_in[1];
    const float* W[20];
    for (int i = 0; i < 20; ++i) W[i] = (const float*)d_in[2 + i];

    const int rows  = in_sizes[0] / 31;          // B*E
    const int tiles = (rows + 15) / 16;

    // Allow >64KB dynamic LDS (deterministic, host-side, capture-safe).
    hipFuncSetAttribute(reinterpret_cast<const void*>(dual_mlp_wmma_kernel),
                        hipFuncAttributeMaxDynamicSharedMemorySize, SMEM_BYTES);

    dim3 grid(1250), block(256);
    dual_mlp_wmma_kernel<<<grid, block, SMEM_BYTES, stream>>>(
        elems, enc,
        W[0], W[1], W[2], W[3], W[4], W[5], W[6], W[7], W[8], W[9],
        W[10], W[11], W[12], W[13], W[14], W[15], W[16], W[17], W[18], W[19],
        (float*)d_out, rows, tiles);
}